// Memory_11828339933191
// MI455X (gfx1250) — compile-verified
//
#include <hip/hip_runtime.h>

#define N_LOC 262144
#define M_VEC 128
#define EPSF  1e-16f

#define K1_BLOCKS   2048   // 128 rows per block
#define K2_BLOCKS   1024   // N / 256
#define K5_BLOCKS   512    // 512 rows per block
#define K5_ROWS     512

typedef __attribute__((ext_vector_type(2))) float v2f;
typedef __attribute__((ext_vector_type(8))) float v8f;

// ---------------- K0: ||key|| ----------------
__global__ void ntm_k0_knorm(const float* __restrict__ key, float* __restrict__ scal) {
    __shared__ float red[128];
    int t = threadIdx.x;                       // 128 threads
    float v = key[t];
    red[t] = v * v;
    __syncthreads();
    for (int off = 64; off > 0; off >>= 1) {
        if (t < off) red[t] += red[t + off];
        __syncthreads();
    }
    if (t == 0) scal[0] = sqrtf(red[0]);
}

// ---------------- K1: scores s_i = beta * cos(mem_i, key); per-block max ----------------
// Default (RT) loads deliberately leave `mem` resident in the 192MB L2 for K5's re-read.
__global__ void ntm_k1_score(const float* __restrict__ mem, const float* __restrict__ key,
                             const float* __restrict__ strength, const float* __restrict__ scal,
                             float* __restrict__ e_arr, float* __restrict__ bmax) {
    int lane = threadIdx.x & 31;
    int wid  = threadIdx.x >> 5;               // 8 waves, wave per row
    float4 k4   = ((const float4*)key)[lane];  // key[4*lane .. +3]
    float knorm = scal[0];
    float beta  = strength[0];
    float wmax  = -__builtin_huge_valf();
    int base = blockIdx.x * 128;
    for (int it = 0; it < 16; ++it) {
        int row = base + it * 8 + wid;
        float4 m4 = ((const float4*)(mem + (size_t)row * M_VEC))[lane];
        float dot = m4.x * k4.x + m4.y * k4.y + m4.z * k4.z + m4.w * k4.w;
        float sq  = m4.x * m4.x + m4.y * m4.y + m4.z * m4.z + m4.w * m4.w;
        for (int off = 16; off > 0; off >>= 1) {
            dot += __shfl_xor(dot, off, 32);
            sq  += __shfl_xor(sq,  off, 32);
        }
        float s = beta * dot / fmaxf(sqrtf(sq) * knorm, EPSF);
        if (lane == 0) e_arr[row] = s;
        wmax = fmaxf(wmax, s);
    }
    __shared__ float red[8];
    if (lane == 0) red[wid] = wmax;
    __syncthreads();
    if (threadIdx.x == 0) {
        float m = red[0];
        for (int i = 1; i < 8; ++i) m = fmaxf(m, red[i]);
        bmax[blockIdx.x] = m;
    }
}

// ---------------- K1b: global max (deterministic tree) ----------------
__global__ void ntm_k1b_max(const float* __restrict__ bmax, float* __restrict__ scal) {
    __shared__ float red[256];
    int t = threadIdx.x;
    float m = -__builtin_huge_valf();
    for (int i = t; i < K1_BLOCKS; i += 256) m = fmaxf(m, bmax[i]);
    red[t] = m;
    __syncthreads();
    for (int off = 128; off > 0; off >>= 1) {
        if (t < off) red[t] = fmaxf(red[t], red[t + off]);
        __syncthreads();
    }
    if (t == 0) scal[1] = red[0];
}

// ---------------- K2: e_i = exp(s_i - smax); per-block sum ----------------
__global__ void ntm_k2_exp(float* __restrict__ e_arr, const float* __restrict__ scal,
                           float* __restrict__ bsum) {
    int i = blockIdx.x * 256 + threadIdx.x;
    float e = expf(e_arr[i] - scal[1]);
    e_arr[i] = e;
    __shared__ float red[256];
    red[threadIdx.x] = e;
    __syncthreads();
    for (int off = 128; off > 0; off >>= 1) {
        if (threadIdx.x < off) red[threadIdx.x] += red[threadIdx.x + off];
        __syncthreads();
    }
    if (threadIdx.x == 0) bsum[blockIdx.x] = red[0];
}

// ---------------- generic deterministic sum of n partials -> *out ----------------
__global__ void ntm_kred_sum(const float* __restrict__ in, int n, float* __restrict__ out) {
    __shared__ float red[256];
    int t = threadIdx.x;
    float s = 0.0f;
    for (int i = t; i < n; i += 256) s += in[i];
    red[t] = s;
    __syncthreads();
    for (int off = 128; off > 0; off >>= 1) {
        if (t < off) red[t] += red[t + off];
        __syncthreads();
    }
    if (t == 0) *out = red[0];
}

// ---------------- K4: interpolate + circular 3-tap correlate + pow; per-block sum ----------------
__global__ void ntm_k4_shiftpow(const float* __restrict__ e_arr, const float* __restrict__ prev,
                                const float* __restrict__ scal, const float* __restrict__ gate_p,
                                const float* __restrict__ shiftw, const float* __restrict__ sharp_p,
                                float* __restrict__ w_out, float* __restrict__ bsum) {
    int i = blockIdx.x * 256 + threadIdx.x;
    float invE = 1.0f / scal[2];
    float g  = gate_p[0];
    float c0 = shiftw[0], c1 = shiftw[1], c2 = shiftw[2];
    float p  = sharp_p[0];
    int im = (i - 1) & (N_LOC - 1);
    int ip = (i + 1) & (N_LOC - 1);
    float gm = g * e_arr[im] * invE + (1.0f - g) * prev[im];
    float gc = g * e_arr[i]  * invE + (1.0f - g) * prev[i];
    float gp = g * e_arr[ip] * invE + (1.0f - g) * prev[ip];
    float sh = c0 * gm + c1 * gc + c2 * gp;     // out[i] = k0*w[i-1]+k1*w[i]+k2*w[i+1]
    float pw = powf(sh, p);
    w_out[i] = pw;
    __shared__ float red[256];
    red[threadIdx.x] = pw;
    __syncthreads();
    for (int off = 128; off > 0; off >>= 1) {
        if (threadIdx.x < off) red[threadIdx.x] += red[threadIdx.x + off];
        __syncthreads();
    }
    if (threadIdx.x == 0) bsum[blockIdx.x] = red[0];
}

// ---------------- K4c: normalize w in place ----------------
__global__ void ntm_k4c_norm(float* __restrict__ w, const float* __restrict__ scal) {
    int i = blockIdx.x * 256 + threadIdx.x;
    w[i] = w[i] / (scal[3] + EPSF);
}

// ---------------- K5: fused read_out (WMMA f32 16x16x4) + erase/add rewrite ----------------
// Block: 256 threads = 8 waves; wave `wid` owns columns [16*wid, 16*wid+16).
// Per iteration the wave consumes a 4-row x 16-col f32 tile of mem:
//   B slot (vgpr v, half h) -> k = v + 2h  -> row i0 + v + 2h
//   A slot (vgpr v, half h) -> k = v + 2h  -> broadcast w[i0 + v + 2h]
// D[m][n] accumulates sum_k w[i0+k]*mem[i0+k][col0+n]; all m rows identical.
// mem reads expected to hit L2 (left warm by K1); new_mem stores are non-temporal
// so the 128MB store stream does not evict mem from L2.
__global__ void ntm_k5_readwrite(const float* __restrict__ mem, const float* __restrict__ w,
                                 const float* __restrict__ erase, const float* __restrict__ addv,
                                 float* __restrict__ new_mem, float* __restrict__ ro_partial) {
    int lane = threadIdx.x & 31;
    int wid  = threadIdx.x >> 5;
    int col0 = wid * 16;
    int half = lane >> 4;
    int n    = lane & 15;
    int col  = col0 + n;
    float ev = erase[col];
    float av = addv[col];
    v8f c = {0.f, 0.f, 0.f, 0.f, 0.f, 0.f, 0.f, 0.f};
    int rbase = blockIdx.x * K5_ROWS;
    int rend  = rbase + K5_ROWS;
    for (int i0 = rbase; i0 < rend; i0 += 4) {
        int rA = i0 + 2 * half;
        float2 wp = *(const float2*)(w + rA);                 // w[rA], w[rA+1]
        const float* mrow = mem + (size_t)rA * M_VEC + col;
        if (i0 + 8 < rend)
            __builtin_prefetch(mem + (size_t)(rA + 8) * M_VEC + col, 0, 3);
        float b0 = mrow[0];
        float b1 = mrow[M_VEC];
        v2f a = {wp.x, wp.y};
        v2f b = {b0, b1};
        c = __builtin_amdgcn_wmma_f32_16x16x4_f32(
                /*neg_a=*/false, a, /*neg_b=*/false, b,
                /*c_mod=*/(short)0, c, /*reuse_a=*/false, /*reuse_b=*/false);
        float* orow = new_mem + (size_t)rA * M_VEC + col;
        float nm0 = b0 * (1.0f - wp.x * ev) + wp.x * av;
        float nm1 = b1 * (1.0f - wp.y * ev) + wp.y * av;
        __builtin_nontemporal_store(nm0, &orow[0]);
        __builtin_nontemporal_store(nm1, &orow[M_VEC]);
    }
    // lanes 0-15 hold D[M=0][N=lane] in c[0]; rows are identical -> take row 0.
    if (lane < 16) ro_partial[blockIdx.x * M_VEC + col0 + n] = c[0];
}

// ---------------- K6: deterministic reduce of read_out partials ----------------
__global__ void ntm_k6_ro(const float* __restrict__ ro_partial, float* __restrict__ ro_out) {
    int t = threadIdx.x;                       // 128 threads
    float s = 0.0f;
    for (int b = 0; b < K5_BLOCKS; ++b) s += ro_partial[b * M_VEC + t];
    ro_out[t] = s;
}

extern "C" void kernel_launch(void* const* d_in, const int* in_sizes, int n_in,
                              void* d_out, int out_size, void* d_ws, size_t ws_size,
                              hipStream_t stream) {
    (void)in_sizes; (void)n_in; (void)out_size; (void)ws_size;
    const float* mem      = (const float*)d_in[0];   // (1,N,M)
    const float* key      = (const float*)d_in[1];   // (M,)
    const float* strength = (const float*)d_in[2];   // (1,)
    const float* gate     = (const float*)d_in[3];   // (1,)
    const float* shiftw   = (const float*)d_in[4];   // (S,)
    const float* sharp    = (const float*)d_in[5];   // (1,)
    const float* prev     = (const float*)d_in[6];   // (N,)
    const float* erase    = (const float*)d_in[7];   // (M,)
    const float* addv     = (const float*)d_in[8];   // (M,)

    float* out    = (float*)d_out;
    float* ro_out = out;                              // [0,128)
    float* newmem = out + M_VEC;                      // [128, 128+N*M)
    float* w_out  = out + M_VEC + (size_t)N_LOC * M_VEC;  // [.., +N)

    float* ws    = (float*)d_ws;
    float* scal  = ws;                 // [0]=||key||, [1]=smax, [2]=Esum, [3]=powsum
    float* e_arr = ws + 16;            // N floats (scores, then exp values)
    float* bmax  = e_arr + N_LOC;      // K1_BLOCKS
    float* bsumE = bmax + K1_BLOCKS;   // K2_BLOCKS
    float* bsumP = bsumE + K2_BLOCKS;  // K2_BLOCKS
    float* ro_p  = bsumP + K2_BLOCKS;  // K5_BLOCKS * 128

    ntm_k0_knorm   <<<1, 128, 0, stream>>>(key, scal);
    ntm_k1_score   <<<K1_BLOCKS, 256, 0, stream>>>(mem, key, strength, scal, e_arr, bmax);
    ntm_k1b_max    <<<1, 256, 0, stream>>>(bmax, scal);
    ntm_k2_exp     <<<K2_BLOCKS, 256, 0, stream>>>(e_arr, scal, bsumE);
    ntm_kred_sum   <<<1, 256, 0, stream>>>(bsumE, K2_BLOCKS, &scal[2]);
    ntm_k4_shiftpow<<<K2_BLOCKS, 256, 0, stream>>>(e_arr, prev, scal, gate, shiftw, sharp,
                                                   w_out, bsumP);
    ntm_kred_sum   <<<1, 256, 0, stream>>>(bsumP, K2_BLOCKS, &scal[3]);
    ntm_k4c_norm   <<<K2_BLOCKS, 256, 0, stream>>>(w_out, scal);
    ntm_k5_readwrite<<<K5_BLOCKS, 256, 0, stream>>>(mem, w_out, erase, addv, newmem, ro_p);
    ntm_k6_ro      <<<1, 128, 0, stream>>>(ro_p, ro_out);
}